// LaneFusionEncoder_40243843564283
// MI455X (gfx1250) — compile-verified
//
#include <hip/hip_runtime.h>
#include <math.h>

// ---------------------------------------------------------------------------
// LaneFusionEncoder for gfx1250 (MI455X).
// B=64, P=256, V=20, C=128, T=64, H=192, DEPTH=3.  N = B*P = 16384 lanes.
// One workgroup (256 threads = 8 wave32) per lane; activations LDS-resident;
// all GEMMs via V_WMMA_F32_16X16X4_F32 (full fp32 precision on matrix pipes).
// ---------------------------------------------------------------------------

#define BPN   16384            // B*P
#define VPTS  20
#define CDIM  128
#define TDIM  64
#define HDIM  192
#define XOUT  (BPN * HDIM)     // 3,145,728 floats of x_result

typedef __attribute__((ext_vector_type(2))) float v2f;
typedef __attribute__((ext_vector_type(8))) float v8f;

__device__ __forceinline__ v8f wmma4(v2f a, v2f b, v8f c) {
  // D(16x16,f32) = A(16x4,f32) * B(4x16,f32) + C
  return __builtin_amdgcn_wmma_f32_16x16x4_f32(
      /*neg_a=*/false, a, /*neg_b=*/false, b,
      /*c_mod=*/(short)0, c, /*reuse_a=*/false, /*reuse_b=*/false);
}

__device__ __forceinline__ float gelu_f(float x) {
  // exact gelu (approximate=False): 0.5*x*(1+erf(x/sqrt(2)))
  return 0.5f * x * (1.0f + erff(x * 0.70710678118654752f));
}

// Generic WMMA GEMM stage: D[M,N] = op( A[M,K] (opt. transposed-in-LDS) * W[K,N] + bias )
// A: LDS (generic ptr), W/bias: global.  One 16x16 tile per wave per iteration.
// Fragment layouts per CDNA5 ISA 7.12.2:
//   A vgpr j, lane l: element (m = l&15, k = kb + j + 2*(l>>4))
//   B vgpr j, lane l: element (k = kb + j + 2*(l>>4), n = l&15)
//   C/D vgpr v, lane l: element (m = v + 8*(l>>4), n = l&15)
template <bool TA, bool TD, bool GELU, bool ACC>
__device__ __forceinline__ void gemm_stage(
    const float* A, int lda,
    const float* __restrict__ W, int ldw,
    const float* __restrict__ bias,
    float* D, int ldd,
    int M, int N, int K,
    int wid, int l15, int hi)
{
  const int NT = N >> 4;
  const int nt = (M >> 4) * NT;
  for (int t = wid; t < nt; t += 8) {          // wave-uniform: EXEC stays all-ones
    const int m0 = (t / NT) << 4;
    const int n0 = (t % NT) << 4;
    v8f acc = {0.f, 0.f, 0.f, 0.f, 0.f, 0.f, 0.f, 0.f};
    for (int kb = 0; kb < K; kb += 4) {
      const int ka = kb + (hi << 1);
      v2f a, b;
      if (TA) {                                // A[m][k] = S[k][m] (free transpose)
        a.x = A[ka * lda + m0 + l15];
        a.y = A[(ka + 1) * lda + m0 + l15];
      } else {
        a.x = A[(m0 + l15) * lda + ka];
        a.y = A[(m0 + l15) * lda + ka + 1];
      }
      b.x = W[ka * ldw + n0 + l15];
      b.y = W[(ka + 1) * ldw + n0 + l15];
      acc = wmma4(a, b, acc);
    }
    const float bv = bias[n0 + l15];
#pragma unroll
    for (int v = 0; v < 8; ++v) {
      float r = acc[v] + bv;
      if (GELU) r = gelu_f(r);
      const int m   = m0 + v + (hi << 3);
      const int idx = TD ? ((n0 + l15) * ldd + m) : (m * ldd + n0 + l15);
      if (ACC) D[idx] += r; else D[idx] = r;
    }
  }
}

// LayerNorm over last dim (C=128) of a (64,C) LDS tile. 4 threads per row.
__device__ __forceinline__ void layernorm64x128(
    const float* S, int lds, float* D, int ldd,
    const float* __restrict__ g, const float* __restrict__ b)
{
  const int tt = threadIdx.x >> 2;   // 0..63 row
  const int q  = threadIdx.x & 3;
  const float* row = S + tt * lds;
  float s = 0.f, s2 = 0.f;
  for (int c = q; c < 128; c += 4) { float v = row[c]; s += v; s2 += v * v; }
  s  += __shfl_xor(s, 1);  s  += __shfl_xor(s, 2);
  s2 += __shfl_xor(s2, 1); s2 += __shfl_xor(s2, 2);
  const float mu   = s * (1.0f / 128.0f);
  const float rstd = rsqrtf(s2 * (1.0f / 128.0f) - mu * mu + 1e-5f);
  float* drow = D + tt * ldd;
  for (int c = q; c < 128; c += 4)
    drow[c] = (row[c] - mu) * rstd * g[c] + b[c];
}

struct LfeParams {
  const float *x, *sl, *hs;
  const float *cpreW1, *cpreb1, *cpreW2, *cpreb2;
  const float *tpreW1, *tpreb1, *tpreW2, *tpreb2;
  const float *n1g[3], *n1b[3], *tW1[3], *tb1[3], *tW2[3], *tb2[3];
  const float *n2g[3], *n2b[3], *cW1[3], *cb1[3], *cW2[3], *cb2[3];
  const float *ng, *nb, *embW1, *embb1, *embW2, *embb2;
  const float *slW, *slb, *unk, *trW, *trb;
  float *out;
};

__global__ __launch_bounds__(256)
void lane_fusion_encoder_kernel(LfeParams P)
{
  // LDS: ~105 KB -> 3 workgroups / WGP (320 KB LDS on CDNA5)
  __shared__ float sA[128 * 68];     // scratch (viewed as stride 132 or 68)
  __shared__ float sB[128 * 68];     // scratch
  __shared__ float sT[64 * 132];     // t tensor, (T=64, C=128) + pad
  __shared__ float sVec[128];
  __shared__ float sLn[128];
  __shared__ float sHid[192];
  __shared__ float sRed[2];
  __shared__ int   sFlag;

  const int n    = blockIdx.x;       // lane id in [0, 16384)
  const int tid  = threadIdx.x;
  const int wid  = tid >> 5;
  const int lane = tid & 31;
  const int hi   = lane >> 4;
  const int l15  = lane & 15;

  const float* xl = P.x + (size_t)n * (VPTS * 12);

  if (tid == 0) sFlag = 0;

  // ---- pos output (independent of the MLP pipeline) ----
  if (tid < 7) {
    const float xy0 = xl[10 * 12 + 0], xy1 = xl[10 * 12 + 1];
    const float hc  = xl[10 * 12 + 2], hsn = xl[10 * 12 + 3];
    const float hd  = atan2f(hsn, hc);
    const float pv[7] = { xy0, xy1, cosf(hd), sinf(hd), 0.f, 0.f, 1.f };
    P.out[XOUT + BPN + (size_t)n * 7 + tid] = pv[tid];
  }
  __syncthreads();

  // ---- load xf (20x8) into sB (stride 12), zero-pad M to 32; valid flag ----
  for (int i = tid; i < 32 * 8; i += 256) {
    const int r = i >> 3, c = i & 7;
    float v = 0.f;
    if (r < VPTS) {
      v = xl[r * 12 + c];
      if (v != 0.f) sFlag = 1;      // benign race: all writers store 1
    }
    sB[r * 12 + c] = v;
  }
  __syncthreads();

  // ---- cpre: (32,8)@(8,128) gelu -> (32,128)@(128,128) ----
  gemm_stage<false, false, true , false>(sB, 12,  P.cpreW1, 128, P.cpreb1, sA, 132, 32, 128, 8,   wid, l15, hi);
  __syncthreads();
  gemm_stage<false, false, false, false>(sA, 132, P.cpreW2, 128, P.cpreb2, sB, 132, 32, 128, 128, wid, l15, hi);
  __syncthreads();
  // ---- tpre: (128,20)[=t0^T]@(20,64) gelu -> (128,64)@(64,64) -> store transposed as t(64,128) ----
  gemm_stage<true , false, true , false>(sB, 132, P.tpreW1, 64,  P.tpreb1, sA, 68,  128, 64, 20,  wid, l15, hi);
  __syncthreads();
  gemm_stage<false, true , false, false>(sA, 68,  P.tpreW2, 64,  P.tpreb2, sT, 132, 128, 64, 64,  wid, l15, hi);
  __syncthreads();

  // ---- 3 mixer blocks ----
  for (int b = 0; b < 3; ++b) {
    layernorm64x128(sT, 132, sA, 132, P.n1g[b], P.n1b[b]);
    __syncthreads();
    // token MLP on transposed view: (128,64)@(64,64) gelu @(64,64); residual-add transposed into t
    gemm_stage<true , false, true , false>(sA, 132, P.tW1[b], 64,  P.tb1[b], sB, 68,  128, 64, 64,  wid, l15, hi);
    __syncthreads();
    gemm_stage<false, true , false, true >(sB, 68,  P.tW2[b], 64,  P.tb2[b], sT, 132, 128, 64, 64,  wid, l15, hi);
    __syncthreads();
    layernorm64x128(sT, 132, sA, 132, P.n2g[b], P.n2b[b]);
    __syncthreads();
    // channel MLP: (64,128)@(128,128) gelu @(128,128); residual-add into t
    gemm_stage<false, false, true , false>(sA, 132, P.cW1[b], 128, P.cb1[b], sB, 132, 64, 128, 128, wid, l15, hi);
    __syncthreads();
    gemm_stage<false, false, false, true >(sB, 132, P.cW2[b], 128, P.cb2[b], sT, 132, 64, 128, 128, wid, l15, hi);
    __syncthreads();
  }

  // ---- mean over T + speed-limit / traffic embeddings ----
  if (tid < 128) {
    const int c = tid;
    float s = 0.f;
    for (int t2 = 0; t2 < 64; ++t2) s += sT[t2 * 132 + c];
    s *= (1.0f / 64.0f);
    const float slv = P.sl[n];
    const bool  hsb = P.hs[n] > 0.5f;
    const float se  = hsb ? (slv * P.slW[c] + P.slb[c]) : P.unk[c];
    float te = P.trb[c];
#pragma unroll
    for (int j = 0; j < 4; ++j) te += xl[8 + j] * P.trW[j * 128 + c];
    sVec[c] = s + se + te;
  }
  __syncthreads();

  // ---- final LayerNorm over 128 ----
  if (wid == 0) {
    float s = 0.f, s2 = 0.f;
    for (int c = lane; c < 128; c += 32) { float v = sVec[c]; s += v; s2 += v * v; }
#pragma unroll
    for (int m = 16; m > 0; m >>= 1) { s += __shfl_xor(s, m); s2 += __shfl_xor(s2, m); }
    if (lane == 0) {
      const float mu = s * (1.0f / 128.0f);
      sRed[0] = mu;
      sRed[1] = rsqrtf(s2 * (1.0f / 128.0f) - mu * mu + 1e-5f);
    }
  }
  __syncthreads();
  if (tid < 128)
    sLn[tid] = (sVec[tid] - sRed[0]) * sRed[1] * P.ng[tid] + P.nb[tid];
  __syncthreads();

  // ---- head: (1,128)@(128,192) gelu -> (1,192)@(192,192); masked store ----
  if (tid < 192) {
    float acc = P.embb1[tid];
    for (int k = 0; k < 128; ++k) acc += sLn[k] * P.embW1[k * 192 + tid];
    sHid[tid] = gelu_f(acc);
  }
  __syncthreads();
  if (tid < 192) {
    float acc = P.embb2[tid];
    for (int k = 0; k < 192; ++k) acc += sHid[k] * P.embW2[k * 192 + tid];
    const float vf = sFlag ? 1.0f : 0.0f;
    P.out[(size_t)n * HDIM + tid] = vf * acc;
  }
  if (tid == 0)
    P.out[XOUT + n] = sFlag ? 0.0f : 1.0f;   // mask_p
}

// ---------------------------------------------------------------------------
// Host launch.  Input order follows jax tree-flatten of setup_inputs():
// x, speed_limit, has_speed_limit, then params leaves with dict keys sorted:
//   blocks[0..2]{cW1,cW2,cb1,cb2,n1b,n1g,n2b,n2g,tW1,tW2,tb1,tb2},
//   cpre_W1, cpre_W2, cpre_b1, cpre_b2, emb_W1, emb_W2, emb_b1, emb_b2,
//   nb, ng, sl_W, sl_b, tpre_W1, tpre_W2, tpre_b1, tpre_b2, tr_W, tr_b, unk
// ---------------------------------------------------------------------------
extern "C" void kernel_launch(void* const* d_in, const int* in_sizes, int n_in,
                              void* d_out, int out_size, void* d_ws, size_t ws_size,
                              hipStream_t stream)
{
  (void)in_sizes; (void)n_in; (void)out_size; (void)d_ws; (void)ws_size;
  LfeParams P;
  P.x  = (const float*)d_in[0];
  P.sl = (const float*)d_in[1];
  P.hs = (const float*)d_in[2];
  for (int b = 0; b < 3; ++b) {
    const int base = 3 + 12 * b;
    P.cW1[b] = (const float*)d_in[base + 0];
    P.cW2[b] = (const float*)d_in[base + 1];
    P.cb1[b] = (const float*)d_in[base + 2];
    P.cb2[b] = (const float*)d_in[base + 3];
    P.n1b[b] = (const float*)d_in[base + 4];
    P.n1g[b] = (const float*)d_in[base + 5];
    P.n2b[b] = (const float*)d_in[base + 6];
    P.n2g[b] = (const float*)d_in[base + 7];
    P.tW1[b] = (const float*)d_in[base + 8];
    P.tW2[b] = (const float*)d_in[base + 9];
    P.tb1[b] = (const float*)d_in[base + 10];
    P.tb2[b] = (const float*)d_in[base + 11];
  }
  P.cpreW1 = (const float*)d_in[39];
  P.cpreW2 = (const float*)d_in[40];
  P.cpreb1 = (const float*)d_in[41];
  P.cpreb2 = (const float*)d_in[42];
  P.embW1  = (const float*)d_in[43];
  P.embW2  = (const float*)d_in[44];
  P.embb1  = (const float*)d_in[45];
  P.embb2  = (const float*)d_in[46];
  P.nb     = (const float*)d_in[47];
  P.ng     = (const float*)d_in[48];
  P.slW    = (const float*)d_in[49];
  P.slb    = (const float*)d_in[50];
  P.tpreW1 = (const float*)d_in[51];
  P.tpreW2 = (const float*)d_in[52];
  P.tpreb1 = (const float*)d_in[53];
  P.tpreb2 = (const float*)d_in[54];
  P.trW    = (const float*)d_in[55];
  P.trb    = (const float*)d_in[56];
  P.unk    = (const float*)d_in[57];
  P.out    = (float*)d_out;

  hipLaunchKernelGGL(lane_fusion_encoder_kernel, dim3(BPN), dim3(256), 0, stream, P);
}